// ProposalNet_26353919328668
// MI455X (gfx1250) — compile-verified
//
#include <hip/hip_runtime.h>

#define KDIM 256
#define NCLS 18
#define OUTC 28   // 3 center + 3 size + 2 heading + 2 obj + 18 sem

typedef __attribute__((ext_vector_type(2))) float v2f;
typedef __attribute__((ext_vector_type(8))) float v8f;

__device__ __forceinline__ unsigned lane_read(unsigned v, int srclane) {
  // gather from srclane (wave32); index in bytes
  return (unsigned)__builtin_amdgcn_ds_bpermute(srclane << 2, (int)v);
}

__device__ __forceinline__ unsigned ballot32(bool p) {
#if __has_builtin(__builtin_amdgcn_ballot_w32)
  return __builtin_amdgcn_ballot_w32(p);
#else
  return (unsigned)__ballot((int)p);
#endif
}

__global__ __launch_bounds__(256, 1) void proposal_suppress_kernel(
    const float* __restrict__ pred_center,
    const float* __restrict__ pred_size,
    const float* __restrict__ pred_heading,
    const float* __restrict__ sem_obj,
    const float* __restrict__ vote_xyz,
    float* __restrict__ out)
{
  __shared__ float cx[KDIM], cy[KDIM], cz[KDIM], n2[KDIM];
  __shared__ float dlx[KDIM], dhx[KDIM], dly[KDIM], dhy[KDIM], dlz[KDIM], dhz[KDIM];
  __shared__ float sc0[KDIM], sc1[KDIM];
  __shared__ unsigned act[KDIM * 8];   // 256 rows x 256-bit "active" matrix

  const int b    = blockIdx.x;
  const int tid  = threadIdx.x;
  const int lane = tid & 31;
  const int wv   = tid >> 5;

  const float* pcb = pred_center  + b * 3 * KDIM;
  const float* psb = pred_size    + b * 3 * KDIM;
  const float* phb = pred_heading + b * 2 * KDIM;
  const float* sob = sem_obj      + b * (2 + NCLS) * KDIM;
  const float* avb = vote_xyz     + b * KDIM * 3;
  float*      outb = out          + b * KDIM * OUTC;

  __builtin_prefetch(sob + tid, 0, 3);   // global_prefetch_b8
  __builtin_prefetch(avb + tid, 0, 3);

  // ---------------- Phase 1: transpose / intervals / stage to LDS ----------
  {
    const int k = tid;
    const float ccx = avb[k*3+0] + pcb[0*KDIM + k];
    const float ccy = avb[k*3+1] + pcb[1*KDIM + k];
    const float ccz = avb[k*3+2] + pcb[2*KDIM + k];
    const float sx  = psb[0*KDIM + k];
    const float sy  = psb[1*KDIM + k];
    const float sz  = psb[2*KDIM + k];
    const float h0  = phb[0*KDIM + k];
    const float h1  = phb[1*KDIM + k];

    cx[k] = ccx; cy[k] = ccy; cz[k] = ccz;
    n2[k] = ccx*ccx + ccy*ccy + ccz*ccz;

    // vectors = size * (h1, 1, h1); high = vectors + center; low = high - |size|
    const float hx = sx*h1 + ccx;
    const float hy = sy    + ccy;
    const float hz = sz*h1 + ccz;
    dlx[k] = hx - fabsf(sx); dhx[k] = hx;
    dly[k] = hy - fabsf(sy); dhy[k] = hy;
    dlz[k] = hz - fabsf(sz); dhz[k] = hz;

    sc0[k] = sob[0*KDIM + k];
    sc1[k] = sob[1*KDIM + k];

    float* o = outb + k * OUTC;
    o[0] = ccx; o[1] = ccy; o[2] = ccz;
    o[3] = sx;  o[4] = sy;  o[5] = sz;
    o[6] = h0;  o[7] = h1;
    #pragma unroll
    for (int c = 0; c < NCLS; ++c) o[10 + c] = sob[(2 + c) * KDIM + k];

    #pragma unroll
    for (int w = 0; w < 8; ++w) act[k * 8 + w] = 0u;
  }
  __syncthreads();

  // ---------------- Phase 2: WMMA Gram matrix -> pairwise "active" bits ----
  // Gram = C (256x3) * C^T via V_WMMA_F32_16X16X4_F32; dist^2 = n2[m]+n2[n]-2*dot.
  {
    const int hl  = lane >> 4;   // 0: K=0,1 (x,y)   1: K=2,3 (z,0)
    const int l16 = lane & 15;
    for (int it = 0; it < 32; ++it) {
      const int t  = wv * 32 + it;          // 256 tiles, 32 per wave
      const int m0 = (t >> 4) << 4;
      const int n0 = (t & 15) << 4;

      v2f av, bv;
      if (hl == 0) {
        av.x = cx[m0 + l16]; av.y = cy[m0 + l16];
        bv.x = cx[n0 + l16]; bv.y = cy[n0 + l16];
      } else {
        av.x = cz[m0 + l16]; av.y = 0.0f;
        bv.x = cz[n0 + l16]; bv.y = 0.0f;
      }

      v8f acc = {};
#if __has_builtin(__builtin_amdgcn_wmma_f32_16x16x4_f32)
      acc = __builtin_amdgcn_wmma_f32_16x16x4_f32(
          false, av, false, bv, (short)0, acc, false, false);
#else
      // fallback: direct dot per C/D element this lane owns
      #pragma unroll
      for (int v = 0; v < 8; ++v) {
        const int m = m0 + v + (hl ? 8 : 0);
        const int n = n0 + l16;
        acc[v] = cx[m]*cx[n] + cy[m]*cy[n] + cz[m]*cz[n];
      }
#endif
      const unsigned word = (unsigned)(n0 >> 5);
      const unsigned sh   = (unsigned)(n0 & 31);   // 0 or 16
      #pragma unroll
      for (int v = 0; v < 8; ++v) {
        const int m = m0 + v + (hl ? 8 : 0);       // C/D layout: lanes 0-15 -> M=v, 16-31 -> M=v+8
        const int n = n0 + l16;
        const bool pred = (n2[m] + n2[n] - 2.0f * acc[v]) < 9.0f;
        const unsigned mk = ballot32(pred);        // low16: row m0+v, high16: row m0+v+8
        if (lane == 0) {
          atomicOr(&act[(m0 + v    ) * 8 + word], (mk & 0xFFFFu) << sh);
          atomicOr(&act[(m0 + v + 8) * 8 + word], (mk >> 16)     << sh);
        }
      }
    }
  }
  __syncthreads();

  // ---------------- Phase 3: suppression scan (wave 0) ----------------------
  // Pass i touches row i many times (1-bit state) and each row j>i once.
  // bit_i transition per step is a {0,1}->{0,1} function; compose with a
  // wave-parallel prefix scan (ds_bpermute), apply swap_j with the exclusive
  // prefix. Representation: (t0,t1) = (f(0), f(1)).
  if (wv == 0) {
    volatile float* vs0 = sc0;
    volatile float* vs1 = sc1;
    for (int i = 0; i < KDIM - 1; ++i) {
      const float di0 = dlx[i], di1 = dhx[i], di2 = dly[i];
      const float di3 = dhy[i], di4 = dlz[i], di5 = dhz[i];
      const float si0 = vs0[i], si1 = vs1[i];   // pass-start scores of row i
      unsigned bit = 0;                          // row-i swap state vs pass start

      for (int jb = i + 1; jb < KDIM; jb += 32) {
        const int  j     = jb + lane;
        const bool valid = (j < KDIM);
        const int  jc    = valid ? j : (KDIM - 1);

        const float dj0 = dlx[jc], dj1 = dhx[jc], dj2 = dly[jc];
        const float dj3 = dhy[jc], dj4 = dlz[jc], dj5 = dhz[jc];
        const float sj0 = vs0[jc], sj1 = vs1[jc];
        const bool  ab  = ((act[i * 8 + (jc >> 5)] >> (jc & 31)) & 1u) != 0u;

        const bool case0 = (dj0<di0) & (dj1>di1) & (dj2<di2) & (dj3>di3) & (dj4<di4) & (dj5>di5);
        const bool case1 = (dj0>di0) & (dj1<di1) & (dj2>di2) & (dj3<di3) & (dj4>di4) & (dj5<di5);
        const bool ovx = ((dj1>di0)&(dj1<di1)) | ((di1>dj0)&(di1<dj1));
        const bool ovy = ((dj3>di2)&(dj3<di3)) | ((di3>dj2)&(di3<dj3));
        const bool ovz = ((dj5>di4)&(dj5<di5)) | ((di5>dj4)&(di5<dj5));
        const bool c2  = (!case0) & (!case1) & (ovx & ovy & ovz);
        const bool active = ab && valid;
        const bool b1 = active && case0;              // swap_i iff zp<zn
        const bool b2 = active && !case0 && case1;    // swap_j iff fp<fn
        const bool b3 = active && c2;                 // swap_i iff zn<fn else swap_j

        unsigned t0 = 0u, t1 = 1u;                    // identity
        if (b1) { t0 = (unsigned)(si0 < si1); t1 = (unsigned)!(si1 < si0); }
        if (b3) { t0 = (unsigned)(si1 < sj1); t1 = (unsigned)!(si0 < sj1); }

        // inclusive prefix composition over lanes (earlier lane applied first)
        unsigned s0 = t0, s1 = t1;
        #pragma unroll
        for (int off = 1; off < 32; off <<= 1) {
          const unsigned p0 = lane_read(s0, lane - off);
          const unsigned p1 = lane_read(s1, lane - off);
          if (lane >= off) {
            const unsigned q0 = p0 ? s1 : s0;
            const unsigned q1 = p1 ? s1 : s0;
            s0 = q0; s1 = q1;
          }
        }
        // exclusive prefix -> bit_i value observed at step (i,j)
        unsigned e0 = lane_read(s0, lane - 1);
        unsigned e1 = lane_read(s1, lane - 1);
        if (lane == 0) { e0 = 0u; e1 = 1u; }
        const unsigned bit_at_j = bit ? e1 : e0;

        bool swapj = false;
        if (b2) swapj = (sj0 < sj1);
        if (b3) { const float zn = bit_at_j ? si0 : si1; swapj = !(zn < sj1); }
        if (swapj && valid) { vs0[j] = sj1; vs1[j] = sj0; }

        // advance chunk-start state by full chunk composition (lane 31 inclusive)
        const unsigned f0 = lane_read(s0, 31);
        const unsigned f1 = lane_read(s1, 31);
        bit = bit ? f1 : f0;
      }
      if (lane == 0 && bit) { vs0[i] = si1; vs1[i] = si0; }
    }
  }
  __syncthreads();

  // ---------------- Phase 4: write suppressed objectness --------------------
  outb[tid * OUTC + 8] = sc0[tid];
  outb[tid * OUTC + 9] = sc1[tid];
}

extern "C" void kernel_launch(void* const* d_in, const int* in_sizes, int n_in,
                              void* d_out, int out_size, void* d_ws, size_t ws_size,
                              hipStream_t stream) {
  (void)in_sizes; (void)n_in; (void)d_ws; (void)ws_size; (void)out_size;
  const float* pred_center  = (const float*)d_in[0];  // (16,3,256)
  const float* pred_size    = (const float*)d_in[1];  // (16,3,256)
  const float* pred_heading = (const float*)d_in[2];  // (16,2,256)
  const float* sem_obj      = (const float*)d_in[3];  // (16,20,256)
  const float* vote_xyz     = (const float*)d_in[4];  // (16,256,3)
  float* out = (float*)d_out;                          // (16,256,28)

  proposal_suppress_kernel<<<dim3(16), dim3(256), 0, stream>>>(
      pred_center, pred_size, pred_heading, sem_obj, vote_xyz, out);
}